// SageConv_37297495998771
// MI455X (gfx1250) — compile-verified
//
#include <hip/hip_runtime.h>
#include <cstdint>

#define NN 100000     // nodes
#define NE 1600000    // edges
#define NG 512        // graphs
#define DE 64         // embed dim
#define DH 128        // hidden dim
#define DO 64         // out dim

typedef __attribute__((ext_vector_type(2))) float v2f;
typedef __attribute__((ext_vector_type(8))) float v8f;

// ---------------- simple fill ----------------
__global__ void k_fill(float* __restrict__ p, int n, float val) {
  int t = blockIdx.x * blockDim.x + threadIdx.x;
  if (t < n) p[t] = val;
}

// ---------------- embedding gather: h0[n,:] = embed[x[n],:] ----------------
__global__ void k_embed(const int* __restrict__ x, const float* __restrict__ emb,
                        float* __restrict__ h0) {
  const int per = DE / 4;                       // 16 float4 per node
  int t = blockIdx.x * blockDim.x + threadIdx.x;
  if (t >= NN * per) return;
  int node = t / per, q = t % per;
  const float4* s = (const float4*)(emb + (size_t)x[node] * DE);
  ((float4*)(h0 + (size_t)node * DE))[q] = s[q];
}

// ---------------- in-degree (float) ----------------
__global__ void k_deg(const int* __restrict__ dst, float* __restrict__ deg) {
  int e = blockIdx.x * blockDim.x + threadIdx.x;
  if (e < NE) atomicAdd(&deg[dst[e]], 1.0f);
}

// ---------------- edge scatter: agg[dst[e],:] += h[src[e],:] ----------------
template<int D>
__global__ void k_scatter(const int* __restrict__ src, const int* __restrict__ dst,
                          const float* __restrict__ h, float* __restrict__ agg) {
  const int per = D / 4;
  int t = blockIdx.x * blockDim.x + threadIdx.x;
  if (t >= NE * per) return;
  int e = t / per, q = t % per;
  float4 v = ((const float4*)(h + (size_t)src[e] * D))[q];
  float* a = agg + (size_t)dst[e] * D + (size_t)q * 4;
  atomicAdd(a + 0, v.x);
  atomicAdd(a + 1, v.y);
  atomicAdd(a + 2, v.z);
  atomicAdd(a + 3, v.w);
}

// ---------------- fused SAGE layer: out = act((agg/deg)@Wl + b + h@Wr) ------
// One wave per 16-node M-tile; V_WMMA_F32_16X16X4_F32 (fp32 matrix pipe).
// A layout (32-bit 16x4): lane = M (lanes 0-15 / 16-31 share M), lane/16
// selects K pair {0,1} vs {2,3}; per-lane float2 holds the two K values.
// B layout mirrored: lane%16 = N, lane/16 selects K pair, v2f holds K,K+1.
// C/D: VGPR m holds row M = m + 8*(lane/16), col N = lane%16.
template<int DIN, int DOUT, bool RELU>
__launch_bounds__(256)
__global__ void k_sage(const float* __restrict__ h_in, const float* __restrict__ agg,
                       const float* __restrict__ deg,
                       const float* __restrict__ Wl, const float* __restrict__ bias,
                       const float* __restrict__ Wr, float* __restrict__ h_out) {
  constexpr int KST = DIN / 4;    // K steps of 4
  constexpr int NT  = DOUT / 16;  // N tiles of 16
  __shared__ float wlds[DIN * DOUT];

  const int lane = threadIdx.x & 31;
  const int wave = threadIdx.x >> 5;
  const int half = lane >> 4;       // which K pair
  const int l16  = lane & 15;
  const int tile = blockIdx.x * 8 + wave;       // 16-node tile (wave-uniform)
  const bool act = (tile * 16) < NN;            // uniform per wave -> EXEC stays full
  const int node = tile * 16 + l16;

  v8f acc[NT] = {};

  // ---- phase 1: stage Wl, accumulate (agg/deg) @ Wl ----
  {
    const float4* g = (const float4*)Wl;
    float4* s = (float4*)wlds;
    for (int i = threadIdx.x; i < DIN * DOUT / 4; i += 256) s[i] = g[i];
  }
  __syncthreads();

  if (act) {
    v2f a[KST];
    float dinv = 1.0f / fmaxf(deg[node], 1.0f);
    const float* row = agg + (size_t)node * DIN + half * 2;
#pragma unroll
    for (int k = 0; k < KST; ++k) {
      v2f v = *(const v2f*)(row + k * 4);
      a[k] = v * dinv;
    }
#pragma unroll
    for (int nt = 0; nt < NT; ++nt) {
#pragma unroll
      for (int k = 0; k < KST; ++k) {
        const float* p = wlds + (size_t)(k * 4 + half * 2) * DOUT + nt * 16 + l16;
        v2f b; b.x = p[0]; b.y = p[DOUT];
        acc[nt] = __builtin_amdgcn_wmma_f32_16x16x4_f32(
            false, a[k], false, b, (short)0, acc[nt], false, false);
      }
    }
  }
  __syncthreads();

  // ---- phase 2: stage Wr, accumulate h @ Wr, epilogue ----
  {
    const float4* g = (const float4*)Wr;
    float4* s = (float4*)wlds;
    for (int i = threadIdx.x; i < DIN * DOUT / 4; i += 256) s[i] = g[i];
  }
  __syncthreads();

  if (act) {
    v2f a[KST];
    const float* row = h_in + (size_t)node * DIN + half * 2;
#pragma unroll
    for (int k = 0; k < KST; ++k) a[k] = *(const v2f*)(row + k * 4);
#pragma unroll
    for (int nt = 0; nt < NT; ++nt) {
#pragma unroll
      for (int k = 0; k < KST; ++k) {
        const float* p = wlds + (size_t)(k * 4 + half * 2) * DOUT + nt * 16 + l16;
        v2f b; b.x = p[0]; b.y = p[DOUT];
        acc[nt] = __builtin_amdgcn_wmma_f32_16x16x4_f32(
            false, a[k], false, b, (short)0, acc[nt], false, false);
      }
    }
    // bias + activation + store
#pragma unroll
    for (int nt = 0; nt < NT; ++nt) {
      float bv = bias[nt * 16 + l16];
#pragma unroll
      for (int m = 0; m < 8; ++m) {
        float r = acc[nt][m] + bv;
        if (RELU) r = fmaxf(r, 0.0f);
        h_out[(size_t)(tile * 16 + half * 8 + m) * DOUT + nt * 16 + l16] = r;
      }
    }
  }
}

// ---------------- per-graph max pool (segment_max) ----------------
__global__ void k_segmax(const float* __restrict__ h, const int* __restrict__ batch,
                         float* __restrict__ out) {
  int t = blockIdx.x * blockDim.x + threadIdx.x;
  if (t >= NN * DO) return;
  int node = t / DO, d = t % DO;
  float v = h[(size_t)node * DO + d];
  float* a = out + (size_t)batch[node] * DO + d;
  // monotone float-max via sign-split integer atomics (out init = -inf)
  if (v >= 0.0f) atomicMax((int*)a, __float_as_int(v));
  else           atomicMin((unsigned int*)a, __float_as_uint(v));
}

extern "C" void kernel_launch(void* const* d_in, const int* in_sizes, int n_in,
                              void* d_out, int out_size, void* d_ws, size_t ws_size,
                              hipStream_t stream) {
  (void)in_sizes; (void)n_in; (void)out_size; (void)ws_size;

  const int*   x   = (const int*)  d_in[0];
  const int*   ei  = (const int*)  d_in[1];   // [2, NE] row-major
  const int*   bat = (const int*)  d_in[2];
  /* d_in[3] = edge_attr: ignored by SAGEConv */
  const float* emb = (const float*)d_in[4];
  const float* W1l = (const float*)d_in[5];
  const float* b1  = (const float*)d_in[6];
  const float* W1r = (const float*)d_in[7];
  const float* W2l = (const float*)d_in[8];
  const float* b2  = (const float*)d_in[9];
  const float* W2r = (const float*)d_in[10];

  const int* src  = ei;
  const int* dstv = ei + NE;

  float* ws   = (float*)d_ws;
  float* h0   = ws;                          // NN*DE
  float* h1   = h0   + (size_t)NN * DE;      // NN*DH
  float* agg1 = h1   + (size_t)NN * DH;      // NN*DE
  float* agg2 = agg1 + (size_t)NN * DE;      // NN*DH
  float* deg  = agg2 + (size_t)NN * DH;      // NN
  float* h2   = h0;                          // h0 dead after layer 1
  float* out  = (float*)d_out;

  const int B = 256;
  const int zn = NN * DE + NN * DH + NN;     // agg1+agg2+deg contiguous

  k_fill<<<(zn + B - 1) / B, B, 0, stream>>>(agg1, zn, 0.0f);
  k_fill<<<(NG * DO + B - 1) / B, B, 0, stream>>>(out, NG * DO,
                                                  -__builtin_huge_valf());
  k_embed<<<(NN * (DE / 4) + B - 1) / B, B, 0, stream>>>(x, emb, h0);
  k_deg<<<(NE + B - 1) / B, B, 0, stream>>>(dstv, deg);

  // layer 1: relu((agg1/deg)@W1l + b1 + h0@W1r) -> h1 [NN, DH]
  k_scatter<DE><<<(NE * (DE / 4) + B - 1) / B, B, 0, stream>>>(src, dstv, h0, agg1);
  k_sage<DE, DH, true><<<(NN / 16 + 7) / 8, B, 0, stream>>>(h0, agg1, deg,
                                                            W1l, b1, W1r, h1);

  // layer 2: (agg2/deg)@W2l + b2 + h1@W2r -> h2 [NN, DO]
  k_scatter<DH><<<(NE * (DH / 4) + B - 1) / B, B, 0, stream>>>(src, dstv, h1, agg2);
  k_sage<DH, DO, false><<<(NN / 16 + 7) / 8, B, 0, stream>>>(h1, agg2, deg,
                                                             W2l, b2, W2r, h2);

  // per-graph max pool
  k_segmax<<<(NN * DO + B - 1) / B, B, 0, stream>>>(h2, bat, out);
}